// MESN_23914377904837
// MI455X (gfx1250) — compile-verified
//
#include <hip/hip_runtime.h>

typedef __attribute__((ext_vector_type(2))) float v2f;
typedef __attribute__((ext_vector_type(8))) float v8f;

// ---------------- degree / norm ----------------

__global__ void init_deg_k(float* __restrict__ deg, int n) {
    int i = blockIdx.x * blockDim.x + threadIdx.x;
    if (i < n) deg[i] = 1.0f;  // self-loop contribution
}

__global__ void count_deg_k(const int* __restrict__ dst, float* __restrict__ deg, int e) {
    int i = blockIdx.x * blockDim.x + threadIdx.x;
    if (i < e) atomicAdd(deg + dst[i], 1.0f);
}

__global__ void dinv_k(float* __restrict__ deg, int n) {
    int i = blockIdx.x * blockDim.x + threadIdx.x;
    if (i < n) {
        float d = deg[i];
        deg[i] = (d > 0.0f) ? rsqrtf(d) : 0.0f;
    }
}

__global__ void norm_k(const int* __restrict__ src, const int* __restrict__ dst,
                       const float* __restrict__ dinv, float* __restrict__ norm, int e) {
    int i = blockIdx.x * blockDim.x + threadIdx.x;
    if (i < e) norm[i] = dinv[src[i]] * dinv[dst[i]];
}

// ---------------- WMMA GEMM: T[n,FOUT] = H[n,FIN] @ W[FIN,FOUT] ----------------
// block = 128 threads = 4 waves; each wave computes one 16-row tile x all FOUT cols.
// FIN is a multiple of 4 (44, 88). FOUT padded to NT*16 with zero columns in LDS.

template <int FIN, int FOUT, int NT>
__launch_bounds__(128)
__global__ void gemm_wmma_k(const float* __restrict__ H, const float* __restrict__ W,
                            float* __restrict__ T, int n) {
    constexpr int NPAD = NT * 16;
    __shared__ float Wl[FIN][NPAD];
    __shared__ float Hl[64][FIN + 1];  // odd stride -> conflict-free A reads

    for (int i = threadIdx.x; i < FIN * NPAD; i += 128) {
        int k = i / NPAD, nn = i - k * NPAD;
        Wl[k][nn] = (nn < FOUT) ? W[k * FOUT + nn] : 0.0f;
    }
    const int rowBase = blockIdx.x * 64;
    for (int i = threadIdx.x; i < 64 * FIN; i += 128) {
        int r = i / FIN, k = i - r * FIN;
        int gr = rowBase + r;
        Hl[r][k] = (gr < n) ? H[(long long)gr * FIN + k] : 0.0f;
    }
    __syncthreads();

    const int wave = threadIdx.x >> 5;
    const int lane = threadIdx.x & 31;
    const int l16  = lane & 15;
    const int half = lane >> 4;           // 0 or 1
    const int mrow = wave * 16 + l16;

    v8f c[NT] = {};
#pragma unroll
    for (int k0 = 0; k0 < FIN; k0 += 4) {
        // A 16x4 layout: lanes0-15 hold K=k0,k0+1; lanes16-31 hold K=k0+2,k0+3
        v2f a;
        a.x = Hl[mrow][k0 + 2 * half + 0];
        a.y = Hl[mrow][k0 + 2 * half + 1];
#pragma unroll
        for (int nt = 0; nt < NT; ++nt) {
            // B 4x16 layout: VGPR p, lane-half h -> row K = k0 + 2*h + p, col N = lane&15
            v2f b;
            b.x = Wl[k0 + 2 * half + 0][nt * 16 + l16];
            b.y = Wl[k0 + 2 * half + 1][nt * 16 + l16];
            c[nt] = __builtin_amdgcn_wmma_f32_16x16x4_f32(
                false, a, false, b, (short)0, c[nt], false, false);
        }
    }

    // C/D layout: VGPR v, lane-half h -> M = v + 8*h, N = lane&15
    const int gr0 = rowBase + wave * 16;
#pragma unroll
    for (int v = 0; v < 8; ++v) {
        int gr = gr0 + v + half * 8;
        if (gr < n) {
#pragma unroll
            for (int nt = 0; nt < NT; ++nt) {
                int col = nt * 16 + l16;
                if (col < FOUT) T[(long long)gr * FOUT + col] = c[nt][v];
            }
        }
    }
}

// ---------------- edge scatter + finalize ----------------

__global__ void zero_k(float* __restrict__ p, int count) {
    int i = blockIdx.x * blockDim.x + threadIdx.x;
    if (i < count) p[i] = 0.0f;
}

template <int F>
__global__ void scatter_k(const float* __restrict__ T, const int* __restrict__ src,
                          const int* __restrict__ dst, const float* __restrict__ norm,
                          float* __restrict__ out, int total) {
    int tid = blockIdx.x * blockDim.x + threadIdx.x;
    if (tid >= total) return;
    int e = tid / F;
    int f = tid - e * F;
    atomicAdd(out + (long long)dst[e] * F + f, T[(long long)src[e] * F + f] * norm[e]);
}

template <int F>
__global__ void finalize_k(float* __restrict__ out, const float* __restrict__ T,
                           const float* __restrict__ dinv, const float* __restrict__ b,
                           int total) {
    int tid = blockIdx.x * blockDim.x + threadIdx.x;
    if (tid >= total) return;
    int i = tid / F;
    int f = tid - i * F;
    float di = dinv[i];
    float v = out[tid] + T[tid] * di * di + b[f];
    out[tid] = fmaxf(v, 0.0f);
}

// ---------------- pooling + heads ----------------

__global__ void pool_k(const float* __restrict__ h, const int* __restrict__ batch,
                       unsigned* __restrict__ pooled, int total) {
    int tid = blockIdx.x * blockDim.x + threadIdx.x;
    if (tid >= total) return;
    int i = tid / 44;
    int f = tid - i * 44;
    // post-ReLU values are >= 0, so uint bit-pattern max == float max; init = 0.0f
    atomicMax(pooled + batch[i] * 44 + f, __float_as_uint(h[tid]));
}

__global__ void mlp_k(const float* __restrict__ feat, const float* __restrict__ Wf1,
                      const float* __restrict__ bf1, const float* __restrict__ Wf2,
                      const float* __restrict__ bf2, float* __restrict__ x2) {
    __shared__ float red[128];
    int g = blockIdx.x, j = threadIdx.x;
    const float* fg = feat + (long long)g * 1019;
    float s = 0.0f;
    for (int k = 0; k < 1019; ++k) s = fmaf(fg[k], Wf1[k * 128 + j], s);
    float hj = fmaxf(s + bf1[j], 0.0f);
    red[j] = hj * Wf2[j];
    __syncthreads();
    for (int off = 64; off > 0; off >>= 1) {
        if (j < off) red[j] += red[j + off];
        __syncthreads();
    }
    if (j == 0) x2[g] = red[0] + bf2[0];
}

__global__ void final_k(const float* __restrict__ pooled, const float* __restrict__ Wg,
                        const float* __restrict__ bg, const float* __restrict__ x2,
                        float* __restrict__ out, int g_cnt) {
    int g = blockIdx.x * blockDim.x + threadIdx.x;
    if (g >= g_cnt) return;
    float s = 0.0f;
    for (int f = 0; f < 44; ++f) s = fmaf(pooled[g * 44 + f], Wg[f], s);
    out[g] = fmaxf(s + bg[0], 0.0f) + x2[g];
}

// ---------------- launcher ----------------

static inline int cdiv(int a, int b) { return (a + b - 1) / b; }

extern "C" void kernel_launch(void* const* d_in, const int* in_sizes, int n_in,
                              void* d_out, int out_size, void* d_ws, size_t ws_size,
                              hipStream_t stream) {
    (void)n_in; (void)out_size; (void)ws_size;
    const float* x       = (const float*)d_in[0];
    const int*   ei      = (const int*)d_in[1];
    const int*   batch   = (const int*)d_in[2];
    const float* feature = (const float*)d_in[3];
    const float* W1 = (const float*)d_in[4];  const float* b1 = (const float*)d_in[5];
    const float* W2 = (const float*)d_in[6];  const float* b2 = (const float*)d_in[7];
    const float* W3 = (const float*)d_in[8];  const float* b3 = (const float*)d_in[9];
    const float* Wg = (const float*)d_in[10]; const float* bg = (const float*)d_in[11];
    const float* Wf1 = (const float*)d_in[12]; const float* bf1 = (const float*)d_in[13];
    const float* Wf2 = (const float*)d_in[14]; const float* bf2 = (const float*)d_in[15];

    const int F = 44;
    const int n = in_sizes[0] / F;
    const int e = in_sizes[1] / 2;
    const int g = in_sizes[3] / 1019;
    const int* src = ei;
    const int* dst = ei + e;

    // workspace carve-out
    char* ws = (char*)d_ws;
    size_t off = 0;
    auto carve = [&](size_t bytes) {
        void* p = ws + off;
        off += (bytes + 255) & ~(size_t)255;
        return p;
    };
    float* dinv   = (float*)carve((size_t)n * 4);
    float* norm   = (float*)carve((size_t)e * 4);
    float* buf0   = (float*)carve((size_t)n * 88 * 4);
    float* buf1   = (float*)carve((size_t)n * 88 * 4);
    float* pooled = (float*)carve((size_t)g * 44 * 4);
    float* x2     = (float*)carve((size_t)g * 4);

    const int B = 256;
    // degree -> dinv -> norm
    init_deg_k<<<cdiv(n, B), B, 0, stream>>>(dinv, n);
    count_deg_k<<<cdiv(e, B), B, 0, stream>>>(dst, dinv, e);
    dinv_k<<<cdiv(n, B), B, 0, stream>>>(dinv, n);
    norm_k<<<cdiv(e, B), B, 0, stream>>>(src, dst, dinv, norm, e);

    const int gB = cdiv(n, 64);

    // Layer 1: 44 -> 44
    gemm_wmma_k<44, 44, 3><<<gB, 128, 0, stream>>>(x, W1, buf0, n);
    zero_k<<<cdiv(n * 44, B), B, 0, stream>>>(buf1, n * 44);
    scatter_k<44><<<cdiv(e * 44, B), B, 0, stream>>>(buf0, src, dst, norm, buf1, e * 44);
    finalize_k<44><<<cdiv(n * 44, B), B, 0, stream>>>(buf1, buf0, dinv, b1, n * 44);

    // Layer 2: 44 -> 88
    gemm_wmma_k<44, 88, 6><<<gB, 128, 0, stream>>>(buf1, W2, buf0, n);
    zero_k<<<cdiv(n * 88, B), B, 0, stream>>>(buf1, n * 88);
    scatter_k<88><<<cdiv(e * 88, B), B, 0, stream>>>(buf0, src, dst, norm, buf1, e * 88);
    finalize_k<88><<<cdiv(n * 88, B), B, 0, stream>>>(buf1, buf0, dinv, b2, n * 88);

    // Layer 3: 88 -> 44
    gemm_wmma_k<88, 44, 3><<<gB, 128, 0, stream>>>(buf1, W3, buf0, n);
    zero_k<<<cdiv(n * 44, B), B, 0, stream>>>(buf1, n * 44);
    scatter_k<44><<<cdiv(e * 44, B), B, 0, stream>>>(buf0, src, dst, norm, buf1, e * 44);
    finalize_k<44><<<cdiv(n * 44, B), B, 0, stream>>>(buf1, buf0, dinv, b3, n * 44);

    // global max pool
    zero_k<<<cdiv(g * 44, B), B, 0, stream>>>(pooled, g * 44);
    pool_k<<<cdiv(n * 44, B), B, 0, stream>>>(buf1, batch, (unsigned*)pooled, n * 44);

    // heads
    mlp_k<<<g, 128, 0, stream>>>(feature, Wf1, bf1, Wf2, bf2, x2);
    final_k<<<cdiv(g, B), B, 0, stream>>>(pooled, Wg, bg, x2, (float*)d_out, g);
}